// LSTMModel_21569325760814
// MI455X (gfx1250) — compile-verified
//
#include <hip/hip_runtime.h>
#include <hip/hip_bf16.h>
#include <math.h>

// ---------------------------------------------------------------------------
// LSTM LM on MI455X (gfx1250, wave32).
//   V=50000, E=512, B=8, T=512.  All math fp32 via V_WMMA_F32_16X16X4_F32:
//   decoder GEMM (0.21 TFLOP) is bound by the 819MB output write (~35us @
//   23.3TB/s), so fp32 matrix math costs nothing vs bf16.
// ---------------------------------------------------------------------------

typedef float v2f __attribute__((ext_vector_type(2)));
typedef float v8f __attribute__((ext_vector_type(8)));

#define E_DIM 512
#define B_DIM 8
#define T_DIM 512
#define V_DIM 50000
#define G_DIM 2048          // 4*E
#define M_DIM 4096          // B*T
#define REC_BLOCKS 32       // persistent recurrence workgroups (16 hidden units each)

static __device__ __forceinline__ v8f wmma_k4(v2f a, v2f b, v8f c) {
  // V_WMMA_F32_16X16X4_F32 : D = A(16x4) * B(4x16) + C, fp32 throughout.
  return __builtin_amdgcn_wmma_f32_16x16x4_f32(false, a, false, b, (short)0, c,
                                               false, false);
}

// ---------------------------------------------------------------------------
// Kernel 1: embedding gather + input projection.
//   xproj[m][n] = dot(embedding[x[m]], w_ih[n]) + b_ih[n] + b_hh[n]
//   m = b*T + t  (4096 rows), n in [0,2048).
// Each wave computes TWO 16x16 C tiles stacked in M (independent WMMA chains,
// shared B fragments).
// ---------------------------------------------------------------------------
__global__ __launch_bounds__(256) void xproj_kernel(
    const int* __restrict__ x, const float* __restrict__ emb,
    const float* __restrict__ w_ih, const float* __restrict__ b_ih,
    const float* __restrict__ b_hh, float* __restrict__ xproj)
{
  const int lane = threadIdx.x & 31;
  const int l16  = lane & 15;
  const int sel  = lane >> 4;                   // 0: K={0,1}, 1: K={2,3}
  const int wave = blockIdx.x * 8 + (threadIdx.x >> 5);

  const int nTilesN = G_DIM / 16;               // 128
  const int m0 = (wave / nTilesN) * 32;         // pair of M tiles
  const int n0 = (wave % nTilesN) * 16;

  const float* Arow0 = emb + (size_t)x[m0 + l16] * E_DIM;       // gathered rows
  const float* Arow1 = emb + (size_t)x[m0 + 16 + l16] * E_DIM;
  const float* Brow  = w_ih + (size_t)(n0 + l16) * E_DIM;       // B = w_ih^T

  v8f c0 = {}, c1 = {};
  #pragma unroll 4
  for (int k0 = 0; k0 < E_DIM; k0 += 4) {
    const int kb = k0 + sel * 2;
    v2f b  = *(const v2f*)(Brow  + kb);
    v2f a0 = *(const v2f*)(Arow0 + kb);
    v2f a1 = *(const v2f*)(Arow1 + kb);
    c0 = wmma_k4(a0, b, c0);
    c1 = wmma_k4(a1, b, c1);
  }

  const float bias = b_ih[n0 + l16] + b_hh[n0 + l16];  // fold both biases
  #pragma unroll
  for (int v = 0; v < 8; ++v) {
    const int r = sel * 8 + v;
    xproj[(size_t)(m0 + r)      * G_DIM + n0 + l16] = c0[v] + bias;
    xproj[(size_t)(m0 + 16 + r) * G_DIM + n0 + l16] = c1[v] + bias;
  }
}

// ---------------------------------------------------------------------------
// Kernel 2: persistent LSTM recurrence. 32 workgroups, each owns 16 hidden
// units (e0..e0+15) => 64 rows of w_hh (~128KB, hot in WGP$/L2 all 512 steps).
// Per step: gates(16x64) = h_prev(16x512, rows 8..15 zero-padded) @ w_hh_subT
// via f32 WMMA (8 waves: 4 N-tiles x 2 K-halves, LDS reduce), then fused
// activations + cell update. h double-buffered in global; software
// release/acquire barrier across the 32 resident workgroups each step.
// ---------------------------------------------------------------------------
__global__ __launch_bounds__(256) void lstm_rec_kernel(
    const float* __restrict__ w_hh, const float* __restrict__ xproj,
    float* __restrict__ hbuf,          // [2][16][512], pre-zeroed
    float* __restrict__ lstm_out,      // [B*T*E]
    unsigned int* __restrict__ bar)    // pre-zeroed
{
  __shared__ float part_s[4][8][16];   // K-half partials
  __shared__ float gates_s[8][64];     // [batch][gate*16 + e_local]
  __shared__ float c_s[8][16];         // persistent cell state

  const int tid   = threadIdx.x;
  const int lane  = tid & 31;
  const int l16   = lane & 15;
  const int sel   = lane >> 4;
  const int wave  = tid >> 5;
  const int gate  = wave & 3;          // i,f,g,o
  const int khalf = wave >> 2;         // K split: [0,256) / [256,512)
  const int e0    = blockIdx.x * 16;

  if (tid < 128) c_s[tid >> 4][tid & 15] = 0.f;
  __syncthreads();

  // B rows for this wave's gate tile: w_hh[gate*512 + e0 + n_local][k]
  const float* Brow = w_hh + (size_t)(gate * E_DIM + e0 + l16) * E_DIM;

  for (int t = 0; t < T_DIM; ++t) {
    const float* Arow = hbuf + (size_t)(t & 1) * 16 * E_DIM + (size_t)l16 * E_DIM;

    v8f c = {};
    const int kstart = khalf * 256;
    #pragma unroll 4
    for (int i = 0; i < 64; ++i) {
      const int kb = kstart + i * 4 + sel * 2;
      v2f a = *(const v2f*)(Arow + kb);
      v2f b = *(const v2f*)(Brow + kb);
      c = wmma_k4(a, b, c);
    }
    if (khalf == 1 && lane < 16) {
      #pragma unroll
      for (int v = 0; v < 8; ++v) part_s[gate][v][l16] = c[v];
    }
    __syncthreads();
    if (khalf == 0 && lane < 16) {
      #pragma unroll
      for (int v = 0; v < 8; ++v)
        gates_s[v][gate * 16 + l16] = c[v] + part_s[gate][v][l16];
    }
    __syncthreads();

    if (tid < 128) {                              // 8 batch x 16 units
      const int b = tid >> 4, e = tid & 15;
      const size_t xb = ((size_t)b * T_DIM + t) * G_DIM + e0 + e;
      float gi = gates_s[b][ 0 + e] + xproj[xb + 0 * E_DIM];
      float gf = gates_s[b][16 + e] + xproj[xb + 1 * E_DIM];
      float gg = gates_s[b][32 + e] + xproj[xb + 2 * E_DIM];
      float go = gates_s[b][48 + e] + xproj[xb + 3 * E_DIM];
      gi = 1.f / (1.f + expf(-gi));
      gf = 1.f / (1.f + expf(-gf));
      gg = tanhf(gg);
      go = 1.f / (1.f + expf(-go));
      const float cn = gf * c_s[b][e] + gi * gg;
      c_s[b][e] = cn;
      const float hn = go * tanhf(cn);
      hbuf[(size_t)((t + 1) & 1) * 16 * E_DIM + (size_t)b * E_DIM + e0 + e] = hn;
      lstm_out[((size_t)b * T_DIM + t) * E_DIM + e0 + e] = hn;
    }

    // device-wide step barrier (all 32 blocks resident by construction)
    __threadfence();
    __syncthreads();
    if (tid == 0) {
      __hip_atomic_fetch_add(bar, 1u, __ATOMIC_RELEASE, __HIP_MEMORY_SCOPE_AGENT);
      const unsigned int target = (unsigned int)REC_BLOCKS * (unsigned int)(t + 1);
      while (__hip_atomic_load(bar, __ATOMIC_ACQUIRE, __HIP_MEMORY_SCOPE_AGENT) < target)
        __builtin_amdgcn_s_sleep(2);
    }
    __syncthreads();
    __threadfence();   // invalidate stale h lines in WGP$ before next step's loads
  }
}

// ---------------------------------------------------------------------------
// Kernel 3: decoder GEMM  out[m][v] = dot(lstm_out[m], dec_w[v]) + dec_b[v].
// M=4096, N=50000 (=3125*16, no tail), K=512. Write-BW bound (~819MB out).
// Two M tiles per wave (dual WMMA chains, shared B fragments).
// ---------------------------------------------------------------------------
__global__ __launch_bounds__(256) void decoder_kernel(
    const float* __restrict__ lstm_out, const float* __restrict__ dec_w,
    const float* __restrict__ dec_b, float* __restrict__ out)
{
  const int lane = threadIdx.x & 31;
  const int l16  = lane & 15;
  const int sel  = lane >> 4;
  const int wave = blockIdx.x * 8 + (threadIdx.x >> 5);

  const int nTilesN = V_DIM / 16;              // 3125
  const int m0 = (wave / nTilesN) * 32;        // pair of M tiles
  const int n0 = (wave % nTilesN) * 16;

  const float* Arow0 = lstm_out + (size_t)(m0 + l16) * E_DIM;
  const float* Arow1 = lstm_out + (size_t)(m0 + 16 + l16) * E_DIM;
  const float* Brow  = dec_w + (size_t)(n0 + l16) * E_DIM;

  v8f c0 = {}, c1 = {};
  #pragma unroll 4
  for (int k0 = 0; k0 < E_DIM; k0 += 4) {
    const int kb = k0 + sel * 2;
    v2f b  = *(const v2f*)(Brow  + kb);
    v2f a0 = *(const v2f*)(Arow0 + kb);
    v2f a1 = *(const v2f*)(Arow1 + kb);
    c0 = wmma_k4(a0, b, c0);
    c1 = wmma_k4(a1, b, c1);
  }

  const float bias = dec_b[n0 + l16];
  #pragma unroll
  for (int v = 0; v < 8; ++v) {
    const int r = sel * 8 + v;
    out[(size_t)(m0 + r)      * V_DIM + n0 + l16] = c0[v] + bias;
    out[(size_t)(m0 + 16 + r) * V_DIM + n0 + l16] = c1[v] + bias;
  }
}

// ---------------------------------------------------------------------------
extern "C" void kernel_launch(void* const* d_in, const int* in_sizes, int n_in,
                              void* d_out, int out_size, void* d_ws, size_t ws_size,
                              hipStream_t stream) {
  const int*   x     = (const int*)d_in[0];
  // d_in[1] = y (unused)
  const float* emb   = (const float*)d_in[2];
  const float* w_ih  = (const float*)d_in[3];
  const float* w_hh  = (const float*)d_in[4];
  const float* b_ih  = (const float*)d_in[5];
  const float* b_hh  = (const float*)d_in[6];
  const float* dec_w = (const float*)d_in[7];
  const float* dec_b = (const float*)d_in[8];
  float* out = (float*)d_out;

  // Workspace layout (~42.1 MB):
  char* ws = (char*)d_ws;
  const size_t xproj_bytes = (size_t)M_DIM * G_DIM * sizeof(float);   // 33.5 MB
  const size_t lstm_bytes  = (size_t)M_DIM * E_DIM * sizeof(float);   //  8.4 MB
  const size_t hbuf_bytes  = (size_t)2 * 16 * E_DIM * sizeof(float);  //  64 KB
  float*        xproj    = (float*)ws;
  float*        lstm_out = (float*)(ws + xproj_bytes);
  float*        hbuf     = (float*)(ws + xproj_bytes + lstm_bytes);
  unsigned int* bar      = (unsigned int*)(ws + xproj_bytes + lstm_bytes + hbuf_bytes);

  // Zero h double-buffer (rows 8..15 stay zero => free M-padding to 16) and
  // the barrier counter. Stream-ordered => graph-capturable.
  hipMemsetAsync(hbuf, 0, hbuf_bytes + 64, stream);

  // 1) embed + input projection: 32768 tiles, 2 M-tiles/wave -> 2048 blocks
  xproj_kernel<<<(M_DIM / 32) * (G_DIM / 16) / 8, 256, 0, stream>>>(
      x, emb, w_ih, b_ih, b_hh, xproj);

  // 2) persistent recurrence: exactly 32 workgroups (one per 16 hidden units)
  lstm_rec_kernel<<<REC_BLOCKS, 256, 0, stream>>>(w_hh, xproj, hbuf, lstm_out, bar);

  // 3) decoder: 128*3125 wave-pairs / 8 waves per block -> 50000 blocks
  decoder_kernel<<<(M_DIM / 32) * (V_DIM / 16) / 8, 256, 0, stream>>>(
      lstm_out, dec_w, dec_b, out);
}